// LightGCN_11458972745825
// MI455X (gfx1250) — compile-verified
//
#include <hip/hip_runtime.h>
#include <hip/hip_bf16.h>
#include <stdint.h>

// LightGCN: 3x COO SpMM (scatter w/ fp32 atomics) + layer mean.
// D is fixed at 64 by the reference model.
#define D_DIM      64
#define CHUNK      1024      // edges staged into LDS per block-iteration
#define SPMM_BLOCK 256       // 8 waves (wave32) = 16 half-waves
#define EW_BLOCK   256

// ---------------------------------------------------------------------------
// CDNA5 async global->LDS copy helpers (gfx1250: tracked by ASYNCcnt)
// ---------------------------------------------------------------------------
__device__ __forceinline__ uint32_t lds_off(const void* p) {
  // Flat LDS addresses carry the LDS byte offset in the low 32 bits.
  return (uint32_t)(uintptr_t)p;
}

__device__ __forceinline__ void async_cp_b128(void* lds, const void* g) {
#if defined(__HIP_DEVICE_COMPILE__) && defined(__gfx1250__)
  asm volatile("global_load_async_to_lds_b128 %0, %1, off"
               :: "v"(lds_off(lds)), "v"(g) : "memory");
#else
  *(int4*)lds = *(const int4*)g;
#endif
}

__device__ __forceinline__ void async_cp_b32(void* lds, const void* g) {
#if defined(__HIP_DEVICE_COMPILE__) && defined(__gfx1250__)
  asm volatile("global_load_async_to_lds_b32 %0, %1, off"
               :: "v"(lds_off(lds)), "v"(g) : "memory");
#else
  *(int*)lds = *(const int*)g;
#endif
}

__device__ __forceinline__ void wait_async_zero() {
#if defined(__HIP_DEVICE_COMPILE__) && defined(__gfx1250__)
  asm volatile("s_wait_asynccnt 0" ::: "memory");
#endif
}

__device__ __forceinline__ void atomic_add_f32(float* p, float v) {
  // Lowers to global_atomic_add_f32 (no return) on gfx1250.
  (void)__hip_atomic_fetch_add(p, v, __ATOMIC_RELAXED, __HIP_MEMORY_SCOPE_AGENT);
}

// ---------------------------------------------------------------------------
// spmm: y[rows[e]] += vals[e] * x[cols[e]]   (D=64 per edge)
// Edge metadata streamed via CDNA5 async global->LDS DMA, double buffered.
// One edge per 16-lane half-wave: float4 gather + 4 fp32 atomics per lane.
// (Defined first so the disasm snippet shows this hot loop.)
// ---------------------------------------------------------------------------
__global__ void __launch_bounds__(SPMM_BLOCK)
lightgcn_spmm_kernel(const int* __restrict__ rows, const int* __restrict__ cols,
                     const float* __restrict__ vals, int nnz,
                     const float* __restrict__ x, float* __restrict__ y,
                     int nchunks) {
  __shared__ __align__(16) int   s_rows[2][CHUNK];
  __shared__ __align__(16) int   s_cols[2][CHUNK];
  __shared__ __align__(16) float s_vals[2][CHUNK];

  const int tid    = threadIdx.x;
  const int halfId = tid >> 4;   // 0..15 half-waves per block
  const int lane16 = tid & 15;

  int chunk = blockIdx.x;
  if (chunk >= nchunks) return;  // uniform over block

  auto issue = [&](int buf, int ck) {
    const int base  = ck * CHUNK;
    int avail = nnz - base;
    if (avail > CHUNK) avail = CHUNK;
    const int idx = tid * 4;                 // 4 elements (16B) per thread
    if (idx + 4 <= avail) {
      async_cp_b128(&s_rows[buf][idx], rows + base + idx);
      async_cp_b128(&s_cols[buf][idx], cols + base + idx);
      async_cp_b128(&s_vals[buf][idx], vals + base + idx);
    } else {
      for (int j = 0; j < 4; ++j) {
        if (idx + j < avail) {
          async_cp_b32(&s_rows[buf][idx + j], rows + base + idx + j);
          async_cp_b32(&s_cols[buf][idx + j], cols + base + idx + j);
          async_cp_b32(&s_vals[buf][idx + j], vals + base + idx + j);
        }
      }
    }
  };

  issue(0, chunk);
  int buf = 0;
  for (; chunk < nchunks; chunk += (int)gridDim.x) {
    wait_async_zero();     // own async portion landed in LDS
    __syncthreads();       // everyone's portion landed; prior buf free for reuse

    const int next = chunk + (int)gridDim.x;
    if (next < nchunks) issue(buf ^ 1, next);   // prefetch next chunk

    const int base = chunk * CHUNK;
    int cnt = nnz - base;
    if (cnt > CHUNK) cnt = CHUNK;

    if (cnt == CHUNK) {
      // Fast path: fixed 64 edges per half-wave, unrolled x4 so four
      // independent 256B gathers are in flight before the atomic burst.
      for (int k = 0; k < CHUNK / 16; k += 4) {
        int   r[4], c[4];
        float v[4];
        float4 xv[4];
#pragma unroll
        for (int u = 0; u < 4; ++u) {
          const int i = halfId + (k + u) * 16;
          r[u] = s_rows[buf][i];
          c[u] = s_cols[buf][i];
          v[u] = s_vals[buf][i];
          xv[u] = *((const float4*)(x + ((size_t)c[u] << 6)) + lane16);
        }
#pragma unroll
        for (int u = 0; u < 4; ++u) {
          float* yp = y + ((size_t)r[u] << 6) + ((size_t)lane16 << 2);
          atomic_add_f32(yp + 0, v[u] * xv[u].x);
          atomic_add_f32(yp + 1, v[u] * xv[u].y);
          atomic_add_f32(yp + 2, v[u] * xv[u].z);
          atomic_add_f32(yp + 3, v[u] * xv[u].w);
        }
      }
    } else {
      for (int i = halfId; i < cnt; i += SPMM_BLOCK / 16) {
        const int   r = s_rows[buf][i];
        const int   c = s_cols[buf][i];
        const float v = s_vals[buf][i];
        const float4 xv = *((const float4*)(x + ((size_t)c << 6)) + lane16);
        float* yp = y + ((size_t)r << 6) + ((size_t)lane16 << 2);
        atomic_add_f32(yp + 0, v * xv.x);
        atomic_add_f32(yp + 1, v * xv.y);
        atomic_add_f32(yp + 2, v * xv.z);
        atomic_add_f32(yp + 3, v * xv.w);
      }
    }
    buf ^= 1;
  }
}

// ---------------------------------------------------------------------------
// init: buf0 = concat(user,item); out = 0.25*buf0; buf1 = 0
// ---------------------------------------------------------------------------
__global__ void __launch_bounds__(EW_BLOCK)
lightgcn_init_kernel(const float* __restrict__ user, const float* __restrict__ item,
                     long userElems, long n4,
                     float* __restrict__ b0, float* __restrict__ b1,
                     float* __restrict__ out) {
  long i = (long)blockIdx.x * blockDim.x + threadIdx.x;
  if (i >= n4) return;
  long e = i * 4;   // userElems is a multiple of 4 (D=64), no straddle
  float4 v;
  if (e < userElems) v = *(const float4*)(user + e);
  else               v = *(const float4*)(item + (e - userElems));
  *(float4*)(b0 + e) = v;
  float4 o = {0.25f * v.x, 0.25f * v.y, 0.25f * v.z, 0.25f * v.w};
  *(float4*)(out + e) = o;
  float4 z = {0.f, 0.f, 0.f, 0.f};
  *(float4*)(b1 + e) = z;
}

// ---------------------------------------------------------------------------
// finish: out += 0.25*src;  optionally zero the next ping-pong buffer
// ---------------------------------------------------------------------------
__global__ void __launch_bounds__(EW_BLOCK)
lightgcn_finish_kernel(float* __restrict__ out, const float* __restrict__ src,
                       float* __restrict__ zbuf, long n4) {
  long i = (long)blockIdx.x * blockDim.x + threadIdx.x;
  if (i >= n4) return;
  long e = i * 4;
  float4 s = *(const float4*)(src + e);
  float4 o = *(float4*)(out + e);
  o.x = fmaf(0.25f, s.x, o.x);
  o.y = fmaf(0.25f, s.y, o.y);
  o.z = fmaf(0.25f, s.z, o.z);
  o.w = fmaf(0.25f, s.w, o.w);
  *(float4*)(out + e) = o;
  if (zbuf) {
    float4 z = {0.f, 0.f, 0.f, 0.f};
    *(float4*)(zbuf + e) = z;
  }
}

// ---------------------------------------------------------------------------
extern "C" void kernel_launch(void* const* d_in, const int* in_sizes, int n_in,
                              void* d_out, int out_size, void* d_ws, size_t ws_size,
                              hipStream_t stream) {
  // setup_inputs order: user_emb, item_emb, adj_vals, adj_rows, adj_cols
  const float* user = (const float*)d_in[0];
  const float* item = (const float*)d_in[1];
  const float* vals = (const float*)d_in[2];
  const int*   rows = (const int*)d_in[3];
  const int*   cols = (const int*)d_in[4];

  const long userElems  = in_sizes[0];
  const long itemElems  = in_sizes[1];
  const int  nnz        = in_sizes[2];
  const long totalElems = userElems + itemElems;   // == out_size == N*64

  float* buf0 = (float*)d_ws;
  float* buf1 = buf0 + totalElems;
  float* out  = (float*)d_out;

  const long n4 = totalElems / 4;
  const int  ewGrid = (int)((n4 + EW_BLOCK - 1) / EW_BLOCK);

  const int nchunks = (nnz + CHUNK - 1) / CHUNK;
  int sg = nchunks < 1536 ? nchunks : 1536;   // grid-stride => double buffering pays off

  // e0
  lightgcn_init_kernel<<<ewGrid, EW_BLOCK, 0, stream>>>(user, item, userElems, n4,
                                                        buf0, buf1, out);
  // layer 1: buf0 -> buf1
  lightgcn_spmm_kernel<<<sg, SPMM_BLOCK, 0, stream>>>(rows, cols, vals, nnz, buf0, buf1, nchunks);
  lightgcn_finish_kernel<<<ewGrid, EW_BLOCK, 0, stream>>>(out, buf1, buf0, n4);
  // layer 2: buf1 -> buf0
  lightgcn_spmm_kernel<<<sg, SPMM_BLOCK, 0, stream>>>(rows, cols, vals, nnz, buf1, buf0, nchunks);
  lightgcn_finish_kernel<<<ewGrid, EW_BLOCK, 0, stream>>>(out, buf0, buf1, n4);
  // layer 3: buf0 -> buf1
  lightgcn_spmm_kernel<<<sg, SPMM_BLOCK, 0, stream>>>(rows, cols, vals, nnz, buf0, buf1, nchunks);
  lightgcn_finish_kernel<<<ewGrid, EW_BLOCK, 0, stream>>>(out, buf1, nullptr, n4);
}